// MEGNet_Node_876173328940
// MI455X (gfx1250) — compile-verified
//
#include <hip/hip_runtime.h>

#define EPSV 1e-5f

typedef float v2f __attribute__((ext_vector_type(2)));
typedef float v8f __attribute__((ext_vector_type(8)));

// ---------------------------------------------------------------- zero scratch
__global__ void zero_f32(float* __restrict__ p, long n) {
    long i = (long)blockIdx.x * blockDim.x + threadIdx.x;
    long st = (long)gridDim.x * blockDim.x;
    for (; i < n; i += st) p[i] = 0.f;
}

// ---------------------------------------------------------------- edge degree
__global__ void edge_count_kernel(const int* __restrict__ src, float* __restrict__ cnt, int E) {
    int i = blockIdx.x * blockDim.x + threadIdx.x;
    int st = gridDim.x * blockDim.x;
    for (; i < E; i += st) atomicAdd(&cnt[src[i]], 1.f);
}

// ------------------------------------------------------- edge_attr scatter-sum
__global__ void edge_scatter_kernel(const int* __restrict__ src,
                                    const float* __restrict__ attr,
                                    float* __restrict__ sums, long total) {
    long i = (long)blockIdx.x * blockDim.x + threadIdx.x;
    long st = (long)gridDim.x * blockDim.x;
    for (; i < total; i += st) {
        int e = (int)(i >> 6);
        int d = (int)(i & 63);
        atomicAdd(&sums[(long)src[e] * 64 + d], attr[i]);
    }
}

// ----------------------------------------------------------------- fused GEMM
// MODE 1: stage comb = [x | sums/cnt | state[batch]]  (KD = 192)
// MODE 2: stage hin * a[c] + b[c]  (fold previous BN)  (KD = 64)
// Output tile per wave: 16 nodes x 16 channels via V_WMMA_F32_16X16X4_F32.
template <int KD, int MODE, bool RELU>
__global__ __launch_bounds__(256) void gemm_bn_kernel(
    const float* __restrict__ x, const float* __restrict__ sums,
    const float* __restrict__ cnt, const float* __restrict__ state,
    const int* __restrict__ batch,
    const float* __restrict__ hin, const float* __restrict__ aff,
    const float* __restrict__ W, const float* __restrict__ bias,
    float* __restrict__ hout, float* __restrict__ gsum, float* __restrict__ gsq,
    int nNodes) {
    constexpr int LD = KD + 4;               // conflict-free LDS pitch
    __shared__ float sC[32 * LD];            // 32-node input slab
    __shared__ float sSum[64];
    __shared__ float sSq[64];

    const int tid = threadIdx.x;
    if (tid < 64) { sSum[tid] = 0.f; sSq[tid] = 0.f; }

    const int wave = tid >> 5;
    const int lane = tid & 31;
    const int l    = lane & 15;
    const int half = lane >> 4;
    const int r    = wave >> 2;              // row tile (0..1)
    const int c    = wave & 3;               // col tile (0..3)
    const int ch   = c * 16 + l;             // output channel of this lane
    const float bval = bias[ch];
    const float* wrow = W + ch * KD + 2 * half;     // B frag: K = k + 2*half

    const int nBlocks = (nNodes + 31) >> 5;
    for (int b = blockIdx.x; b < nBlocks; b += gridDim.x) {
        const int base = b << 5;

        if (MODE == 1) {
            for (int i = tid; i < 32 * KD; i += 256) {
                int row = i / KD, col = i - row * KD;
                int node = base + row;
                float v = 0.f;
                if (node < nNodes) {
                    if (col < 64) {
                        v = x[node * 64 + col];
                    } else if (col < 128) {
                        float cn = fmaxf(cnt[node], 1.f);
                        v = sums[node * 64 + (col - 64)] / cn;
                    } else {
                        v = state[batch[node] * 64 + (col - 128)];
                    }
                }
                sC[row * LD + col] = v;
            }
        } else {
            for (int i = tid; i < 32 * 64; i += 256) {
                int row = i >> 6, col = i & 63;
                int node = base + row;
                float v = 0.f;
                if (node < nNodes)
                    v = fmaf(hin[node * 64 + col], aff[col], aff[64 + col]);
                sC[row * LD + col] = v;
            }
        }
        __syncthreads();

        v8f acc = {0.f, 0.f, 0.f, 0.f, 0.f, 0.f, 0.f, 0.f};
        const float* arow = &sC[(r * 16 + l) * LD + 2 * half];
#pragma unroll 4
        for (int k = 0; k < KD; k += 4) {
            v2f av = *(const v2f*)(arow + k);   // A: 16x4 f32 fragment
            v2f bv = *(const v2f*)(wrow + k);   // B: 4x16 f32 fragment (W^T)
            acc = __builtin_amdgcn_wmma_f32_16x16x4_f32(
                false, av, false, bv, (short)0, acc, false, false);
        }

        float s = 0.f, q = 0.f;
        const int nodeBase = base + r * 16 + half * 8;   // D: M = i + 8*half
        if (base + 32 <= nNodes) {
            // fast path: whole 32-node block in range (always taken for N%32==0)
            float* orow = hout + (long)nodeBase * 64 + ch;
#pragma unroll
            for (int i = 0; i < 8; i++) {
                float v = acc[i] + bval;
                if (RELU) v = fmaxf(v, 0.f);
                orow[i * 64] = v;
                s += v;
                q += v * v;
            }
        } else {
#pragma unroll
            for (int i = 0; i < 8; i++) {
                int node = nodeBase + i;
                float v = acc[i] + bval;
                if (RELU) v = fmaxf(v, 0.f);
                if (node < nNodes) {
                    hout[node * 64 + ch] = v;
                    s += v;
                    q += v * v;
                }
            }
        }
        atomicAdd(&sSum[ch], s);
        atomicAdd(&sSq[ch], q);
        __syncthreads();
    }

    if (tid < 64) {
        atomicAdd(&gsum[tid], sSum[tid]);
        atomicAdd(&gsq[tid], sSq[tid]);
    }
}

// ------------------------------------------- BN stats -> per-channel affine
__global__ void bn_affine_kernel(const float* __restrict__ gsum,
                                 const float* __restrict__ gsq,
                                 const float* __restrict__ gamma,
                                 const float* __restrict__ beta,
                                 float* __restrict__ ab, float invN) {
    int cdx = threadIdx.x;
    if (cdx < 64) {
        float mean = gsum[cdx] * invN;
        float var  = gsq[cdx] * invN - mean * mean;   // biased variance
        float a    = gamma[cdx] * rsqrtf(var + EPSV);
        ab[cdx]      = a;
        ab[64 + cdx] = fmaf(-mean, a, beta[cdx]);
    }
}

// ----------------------------------------------------- final BN application
__global__ void bn_apply_kernel(const float* __restrict__ h,
                                const float* __restrict__ ab,
                                float* __restrict__ out, long total) {
    long i = (long)blockIdx.x * blockDim.x + threadIdx.x;
    long st = (long)gridDim.x * blockDim.x;
    for (; i < total; i += st) {
        int cdx = (int)(i & 63);
        out[i] = fmaf(h[i], ab[cdx], ab[64 + cdx]);
    }
}

extern "C" void kernel_launch(void* const* d_in, const int* in_sizes, int n_in,
                              void* d_out, int out_size, void* d_ws, size_t ws_size,
                              hipStream_t stream) {
    const float* x     = (const float*)d_in[0];
    const int*   eidx  = (const int*)d_in[1];     // [2,E], row 0 = src
    const float* eattr = (const float*)d_in[2];   // [E,64]
    const float* state = (const float*)d_in[3];   // [B,64]
    const int*   batch = (const int*)d_in[4];     // [N]
    const float* W1 = (const float*)d_in[5];
    const float* b1 = (const float*)d_in[6];
    const float* g1 = (const float*)d_in[7];
    const float* be1 = (const float*)d_in[8];
    const float* W2 = (const float*)d_in[9];
    const float* b2 = (const float*)d_in[10];
    const float* g2 = (const float*)d_in[11];
    const float* be2 = (const float*)d_in[12];
    const float* W3 = (const float*)d_in[13];
    const float* b3 = (const float*)d_in[14];
    const float* g3 = (const float*)d_in[15];
    const float* be3 = (const float*)d_in[16];

    const int N = in_sizes[4];
    const int E = in_sizes[2] / 64;

    // ws layout (floats): sums[N*64] | cnt[N] | stats[6*64] | ab[6*64] | h[N*64]
    float* sums  = (float*)d_ws;
    float* cnt   = sums + (long)N * 64;
    float* stats = cnt + N;
    float* ab    = stats + 384;
    float* h     = ab + 384;

    // zero sums + cnt + stats (ab and h are fully overwritten)
    zero_f32<<<2048, 256, 0, stream>>>(sums, (long)N * 64 + N + 384);

    // scatter-mean numerator/denominator
    edge_count_kernel<<<1024, 256, 0, stream>>>(eidx, cnt, E);
    edge_scatter_kernel<<<4096, 256, 0, stream>>>(eidx, eattr, sums, (long)E * 64);

    // layer 1: comb @ W1^T + b1, relu; stats1
    gemm_bn_kernel<192, 1, true><<<768, 256, 0, stream>>>(
        x, sums, cnt, state, batch, nullptr, nullptr,
        W1, b1, h, stats + 0, stats + 64, N);
    bn_affine_kernel<<<1, 64, 0, stream>>>(stats + 0, stats + 64, g1, be1, ab + 0, 1.f / (float)N);

    // layer 2: (BN1 folded) h @ W2^T + b2, relu; stats2  (in-place h, row-local)
    gemm_bn_kernel<64, 2, true><<<768, 256, 0, stream>>>(
        x, sums, cnt, state, batch, h, ab + 0,
        W2, b2, h, stats + 128, stats + 192, N);
    bn_affine_kernel<<<1, 64, 0, stream>>>(stats + 128, stats + 192, g2, be2, ab + 128, 1.f / (float)N);

    // layer 3: (BN2 folded) h @ W3^T + b3 (no relu); stats3
    gemm_bn_kernel<64, 2, false><<<768, 256, 0, stream>>>(
        x, sums, cnt, state, batch, h, ab + 128,
        W3, b3, h, stats + 256, stats + 320, N);
    bn_affine_kernel<<<1, 64, 0, stream>>>(stats + 256, stats + 320, g3, be3, ab + 256, 1.f / (float)N);

    // final BN3
    bn_apply_kernel<<<2048, 256, 0, stream>>>(h, ab + 256, (float*)d_out, (long)N * 64);
}